// FindSimilarIntentSess_24429773980360
// MI455X (gfx1250) — compile-verified
//
#include <hip/hip_runtime.h>
#include <math.h>

typedef __attribute__((ext_vector_type(2))) float v2f;
typedef __attribute__((ext_vector_type(8))) float v8f;

#define HDIM   100
#define TILE   16
#define KSTEPS 25          // HDIM / 4
#define WAVES  4
#define NTHR   (WAVES * 32)

// ---------------- kernel 1: reciprocal row norms -------------------------
__global__ void rnorm_kernel(const float* __restrict__ E,
                             float* __restrict__ rn, int n) {
    int i = blockIdx.x * blockDim.x + threadIdx.x;
    if (i >= n) return;
    const float* row = E + (size_t)i * HDIM;
    float s = 0.f;
#pragma unroll 4
    for (int c = 0; c < HDIM; ++c) { float x = row[c]; s = fmaf(x, x, s); }
    s += 1e-6f * (float)HDIM;          // sum(x*x + 1e-6) over HDIM elems
    rn[i] = 1.0f / sqrtf(s);
}

// ---------------- kernel 2: fused cos-sim / softmax / top5 / gather ------
__launch_bounds__(NTHR)
__global__ void neighbor_kernel(const float* __restrict__ E,
                                const float* __restrict__ rn,
                                float* __restrict__ out, int n) {
    const int i0   = blockIdx.x * TILE;
    const int tid  = threadIdx.x;
    const int lane = tid & 31;
    const int wave = tid >> 5;
    const int lm   = lane & 15;   // M (A) / N (B) within the 16x16 tile
    const int lh   = lane >> 4;   // selects K-pair {0,1} vs {2,3}
    const int ntiles = n / TILE;

    // Preload A fragments for this row-tile (reused for every column tile)
    v2f afrag[KSTEPS];
    {
        const float* arow = E + (size_t)(i0 + lm) * HDIM;
#pragma unroll
        for (int kk = 0; kk < KSTEPS; ++kk)
            afrag[kk] = *(const v2f*)(arow + 4 * kk + 2 * lh);
    }
    float rni[8];
#pragma unroll
    for (int r = 0; r < 8; ++r) rni[r] = rn[i0 + 8 * lh + r];

    // per-lane online-softmax + top-5 state (8 rows per lane)
    float m[8], s[8], tv[8][5];
    int   ti[8][5];
#pragma unroll
    for (int r = 0; r < 8; ++r) {
        m[r] = -INFINITY; s[r] = 0.f;
#pragma unroll
        for (int k = 0; k < 5; ++k) { tv[r][k] = -INFINITY; ti[r][k] = 0; }
    }

    for (int jt = wave; jt < ntiles; jt += WAVES) {
        const int j0 = jt * TILE;
        const float* brow = E + (size_t)(j0 + lm) * HDIM;
        v8f acc = {};
#pragma unroll
        for (int kk = 0; kk < KSTEPS; ++kk) {
            v2f b = *(const v2f*)(brow + 4 * kk + 2 * lh);
            acc = __builtin_amdgcn_wmma_f32_16x16x4_f32(
                false, afrag[kk], false, b, (short)0, acc, false, false);
        }
        const float rnj = rn[j0 + lm];
        const int   j   = j0 + lm;
#pragma unroll
        for (int r = 0; r < 8; ++r) {
            float c = acc[r] * rni[r] * rnj;          // cosine similarity
            // online softmax (one exp per element)
            float d = c - m[r];
            float e = __expf(-fabsf(d));
            bool  g = d > 0.f;
            s[r] = g ? fmaf(s[r], e, 1.0f) : (s[r] + e);
            m[r] = g ? c : m[r];
            // branchless sorted top-5 insert
            float cv = c; int ci = j;
#pragma unroll
            for (int k = 0; k < 5; ++k) {
                bool  sw = cv > tv[r][k];
                float ov = tv[r][k]; int oi = ti[r][k];
                tv[r][k] = sw ? cv : ov;  ti[r][k] = sw ? ci : oi;
                cv       = sw ? ov : cv;  ci       = sw ? oi : ci;
            }
        }
    }

    // ---- cross-lane / cross-wave merge through LDS ----
    __shared__ float lM[TILE][WAVES * 16];
    __shared__ float lS[TILE][WAVES * 16];
    __shared__ float lV[TILE][WAVES * 16][5];
    __shared__ int   lI[TILE][WAVES * 16][5];
    __shared__ float fw[TILE][5];
    __shared__ int   fi[TILE][5];

    const int slot = wave * 16 + lm;
#pragma unroll
    for (int r = 0; r < 8; ++r) {
        int row = 8 * lh + r;
        lM[row][slot] = m[r];
        lS[row][slot] = s[r];
#pragma unroll
        for (int k = 0; k < 5; ++k) { lV[row][slot][k] = tv[r][k]; lI[row][slot][k] = ti[r][k]; }
    }
    __syncthreads();

    if (tid < TILE) {
        const int row = tid;
        float M = -INFINITY;
        for (int p = 0; p < WAVES * 16; ++p) M = fmaxf(M, lM[row][p]);
        float S = 0.f;
        for (int p = 0; p < WAVES * 16; ++p) S += lS[row][p] * __expf(lM[row][p] - M);

        float v[5]; int id[5];
#pragma unroll
        for (int k = 0; k < 5; ++k) { v[k] = -INFINITY; id[k] = 0; }
        for (int p = 0; p < WAVES * 16; ++p) {
#pragma unroll
            for (int q = 0; q < 5; ++q) {
                float cv = lV[row][p][q]; int ci = lI[row][p][q];
#pragma unroll
                for (int k = 0; k < 5; ++k) {
                    bool  sw = cv > v[k];
                    float ov = v[k]; int oi = id[k];
                    v[k] = sw ? cv : ov; id[k] = sw ? ci : oi;
                    cv   = sw ? ov : cv; ci    = sw ? oi : ci;
                }
            }
        }
        // first softmax values for the top-5, then second softmax over them
        float p0 = __expf(v[0] - M) / S;   // largest probability
        float z = 0.f, w[5];
#pragma unroll
        for (int k = 0; k < 5; ++k) {
            float pk = __expf(v[k] - M) / S;
            float ek = __expf(pk - p0);
            w[k] = ek; z += ek;
        }
        float rz = 1.0f / z;
#pragma unroll
        for (int k = 0; k < 5; ++k) { fw[row][k] = w[k] * rz; fi[row][k] = id[k]; }
    }
    __syncthreads();

    // weighted gather of neighbor embeddings: 16 rows x 100 cols
    for (int e = tid; e < TILE * HDIM; e += NTHR) {
        int row = e / HDIM, col = e % HDIM;
        float a = 0.f;
#pragma unroll
        for (int k = 0; k < 5; ++k)
            a = fmaf(fw[row][k], E[(size_t)fi[row][k] * HDIM + col], a);
        out[(size_t)(i0 + row) * HDIM + col] = a;
    }
}

// ---------------- launcher ----------------------------------------------
extern "C" void kernel_launch(void* const* d_in, const int* in_sizes, int n_in,
                              void* d_out, int out_size, void* d_ws, size_t ws_size,
                              hipStream_t stream) {
    const float* E  = (const float*)d_in[0];
    float*       o  = (float*)d_out;
    float*       rn = (float*)d_ws;              // n floats of scratch
    const int n = in_sizes[0] / HDIM;            // 16384

    rnorm_kernel<<<(n + 255) / 256, 256, 0, stream>>>(E, rn, n);
    neighbor_kernel<<<n / TILE, NTHR, 0, stream>>>(E, rn, o, n);
}